// LOGG3D_71236327571639
// MI455X (gfx1250) — compile-verified
//
#include <hip/hip_runtime.h>
#include <hip/hip_bf16.h>

typedef float v2f __attribute__((ext_vector_type(2)));
typedef float v4f __attribute__((ext_vector_type(4)));
typedef float v8f __attribute__((ext_vector_type(8)));

#define NPTS  256000
#define BATCH 64
#define DIM   256
#define OUTD  128
#define RPB   256          // rows per block in segment-max pass
#define EPS_BN 1e-5f

// ---------------- Kernel 0: zero the pooled output (atomic-max target) -----
__global__ void LOGG3D_zero_pooled(float* __restrict__ pooled) {
    pooled[blockIdx.x * 256 + threadIdx.x] = 0.0f;
}

// ---------------- Kernel 1: streaming segment max --------------------------
// Bandwidth-bound: 262MB of features, read once with b128 non-temporal loads.
// segment_ids are sorted, so each block sees only 1-2 segments; we keep a
// running max in registers (init 0 == the zero-padding max) and flush with
// integer atomic max (valid because flushed values are always >= 0.0f, whose
// IEEE bit patterns order identically to signed ints).
__global__ __launch_bounds__(256) void LOGG3D_seg_max(
    const float* __restrict__ feats, const int* __restrict__ seg,
    int* __restrict__ pooled_bits) {
    __shared__ int sseg[RPB];
    const int t = threadIdx.x;
    const int row0 = blockIdx.x * RPB;
    sseg[t] = seg[row0 + t];
    __syncthreads();

    const int col = (t & 63) << 2;   // 4 columns per thread (float4)
    const int rl  = t >> 6;          // row lane 0..3 (uniform per wave32)
    v4f acc = {0.f, 0.f, 0.f, 0.f};
    int cur = -1;

    for (int i = rl; i < RPB; i += 4) {
        const int s = sseg[i];               // wave-uniform
        if (s != cur) {                      // wave-uniform branch
            if (cur >= 0) {
                int* p = pooled_bits + cur * DIM + col;
                atomicMax(p + 0, __float_as_int(acc.x));
                atomicMax(p + 1, __float_as_int(acc.y));
                atomicMax(p + 2, __float_as_int(acc.z));
                atomicMax(p + 3, __float_as_int(acc.w));
            }
            cur = s;
            acc = (v4f){0.f, 0.f, 0.f, 0.f};
        }
        const v4f v = __builtin_nontemporal_load(
            (const v4f*)(feats + (size_t)(row0 + i) * DIM + col));
        acc.x = fmaxf(acc.x, v.x);
        acc.y = fmaxf(acc.y, v.y);
        acc.z = fmaxf(acc.z, v.z);
        acc.w = fmaxf(acc.w, v.w);
    }
    if (cur >= 0) {
        int* p = pooled_bits + cur * DIM + col;
        atomicMax(p + 0, __float_as_int(acc.x));
        atomicMax(p + 1, __float_as_int(acc.y));
        atomicMax(p + 2, __float_as_int(acc.z));
        atomicMax(p + 3, __float_as_int(acc.w));
    }
}

// ---------------- Kernel 2: fused MLP head (WMMA f32 16x16x4) --------------
// h = pooled @ W1 ; BN(train stats, axis 0) ; ReLU ; proj = h @ W2 + b2
// One 256-thread block (8 waves). h[64][256] resides in LDS (64KB).
//
// V_WMMA_F32_16X16X4_F32 lane layouts (ISA 7.12.2):
//   A 16x4 : lane l holds A[l%16][kb + (l/16)*2 + {0,1}] in 2 VGPRs
//   B 4x16 : lane l holds B[kb + (l/16)*2 + {0,1}][l%16] in 2 VGPRs
//   C 16x16: VGPR v of lane l holds C[v + (l/16)*8][l%16]
__global__ __launch_bounds__(256) void LOGG3D_mlp(
    const float* __restrict__ pooled, const float* __restrict__ W1,
    const float* __restrict__ gamma, const float* __restrict__ beta,
    const float* __restrict__ W2, const float* __restrict__ b2,
    float* __restrict__ proj) {
    __shared__ float h[BATCH * DIM];     // 64KB of the 320KB WGP LDS

    const int lane  = threadIdx.x & 31;
    const int wave  = threadIdx.x >> 5;  // 0..7
    const int lrow  = lane & 15;         // M (A) / N (B,C) index within tile
    const int lhalf = lane >> 4;         // half-wave select

    // ---- GEMM1: 4x16 = 64 output tiles, 8 per wave ----
    for (int tile = wave; tile < 64; tile += 8) {
        const int mt = tile >> 4, nt = tile & 15;
        const int arow = mt * 16 + lrow;
        const int bcol = nt * 16 + lrow;
        v8f c = {};
        for (int kb = 0; kb < DIM; kb += 4) {
            const int ka = kb + lhalf * 2;
            v2f a = *(const v2f*)(pooled + arow * DIM + ka);
            v2f b;
            b.x = W1[ka * DIM + bcol];
            b.y = W1[(ka + 1) * DIM + bcol];
            c = __builtin_amdgcn_wmma_f32_16x16x4_f32(
                false, a, false, b, (short)0, c, false, false);
        }
#pragma unroll
        for (int v = 0; v < 8; ++v)
            h[(mt * 16 + v + lhalf * 8) * DIM + bcol] = c[v];
    }
    __syncthreads();

    // ---- BatchNorm (training stats over 64 rows) + ReLU, in place ----
    {
        const int d = threadIdx.x;          // each thread owns one column
        float s = 0.f, s2 = 0.f;
#pragma unroll 8
        for (int r = 0; r < BATCH; ++r) {
            const float v = h[r * DIM + d];
            s += v; s2 += v * v;
        }
        const float mean = s * (1.0f / BATCH);
        const float var  = s2 * (1.0f / BATCH) - mean * mean;
        const float sc   = rsqrtf(var + EPS_BN) * gamma[d];
        const float bt   = beta[d];
#pragma unroll 8
        for (int r = 0; r < BATCH; ++r) {
            const float v = (h[r * DIM + d] - mean) * sc + bt;
            h[r * DIM + d] = fmaxf(v, 0.f);
        }
    }
    __syncthreads();

    // ---- GEMM2: 4x8 = 32 output tiles, 4 per wave; A from LDS ----
    for (int tile = wave; tile < 32; tile += 8) {
        const int mt = tile >> 3, nt = tile & 7;
        const int arow = mt * 16 + lrow;
        const int bcol = nt * 16 + lrow;
        v8f c = {};
        for (int kb = 0; kb < DIM; kb += 4) {
            const int ka = kb + lhalf * 2;
            v2f a = *(const v2f*)(&h[arow * DIM + ka]);
            v2f b;
            b.x = W2[ka * OUTD + bcol];
            b.y = W2[(ka + 1) * OUTD + bcol];
            c = __builtin_amdgcn_wmma_f32_16x16x4_f32(
                false, a, false, b, (short)0, c, false, false);
        }
        const float bias = b2[bcol];
#pragma unroll
        for (int v = 0; v < 8; ++v)
            proj[(mt * 16 + v + lhalf * 8) * OUTD + bcol] = c[v] + bias;
    }
}

extern "C" void kernel_launch(void* const* d_in, const int* in_sizes, int n_in,
                              void* d_out, int out_size, void* d_ws, size_t ws_size,
                              hipStream_t stream) {
    const float* feats = (const float*)d_in[0];   // [256000, 256]
    const int*   seg   = (const int*)  d_in[1];   // [256000] sorted
    const float* W1    = (const float*)d_in[2];   // [256, 256]
    const float* gamma = (const float*)d_in[3];   // [256]
    const float* beta  = (const float*)d_in[4];   // [256]
    const float* W2    = (const float*)d_in[5];   // [256, 128]
    const float* b2    = (const float*)d_in[6];   // [128]

    float* out    = (float*)d_out;
    float* pooled = out;                          // first 64*256
    float* proj   = out + BATCH * DIM;            // next 64*128

    LOGG3D_zero_pooled<<<(BATCH * DIM) / 256, 256, 0, stream>>>(pooled);
    LOGG3D_seg_max<<<NPTS / RPB, 256, 0, stream>>>(feats, seg, (int*)pooled);
    LOGG3D_mlp<<<1, 256, 0, stream>>>(pooled, W1, gamma, beta, W2, b2, proj);
}